// MCCP_82660940579044
// MI455X (gfx1250) — compile-verified
//
#include <hip/hip_runtime.h>

typedef float v2f __attribute__((ext_vector_type(2)));
typedef float v8f __attribute__((ext_vector_type(8)));

#define B_    512
#define L_    32768
#define NC_   16
#define ND_   8
#define R_    512
#define S_    256
#define W_    128
#define CD_   128
#define KC_   16           // k-floats staged per half; each chunk covers k0 and k0+256
#define NCH_  (R_ / (2 * KC_))   // 16 chunks
#define XSTR_ 20           // padded LDS row stride (dwords): 16 data + 4 pad
#define PSTR_ 130
#define EPS_  1e-4f

// LDS layout (bytes), double-buffered staging:
//   buf{0,1}: xs[128][20] (10240) + wl[2][128][20] (20480)  => 30720 each
//   sig[16][8][8] at tail. Epilogue proj/partbuf overlay dead staging space.
#define XS_BYTES   (W_ * XSTR_ * 4)          // 10240
#define WL_BYTES   (2 * CD_ * XSTR_ * 4)     // 20480
#define BUF_BYTES  (XS_BYTES + WL_BYTES)     // 30720
#define SIG_OFF    (2 * BUF_BYTES)           // 61440
#define SMEM_BYTES (SIG_OFF + NC_ * ND_ * ND_ * 4)  // 65536
#define PART_OFF   (64 * PSTR_ * 4)          // 33280 (after proj overlay)

// async copy: one b128 per lane, global -> LDS, tracked by ASYNCcnt
#define AISSUE(ldsoff, gptr)                                                 \
  asm volatile("global_load_async_to_lds_b128 %0, %1, off"                   \
               :: "v"(ldsoff), "v"(gptr) : "memory")

// ---------------------------------------------------------------------------
// Kernel 1: wc[c] = W_c W_c^T (8x8), sigma[c] = inv(wc + eps*I). Tiny.
// ---------------------------------------------------------------------------
__global__ __launch_bounds__(128) void mccp_prep(const float* __restrict__ weight,
                                                 float* __restrict__ sigma) {
  __shared__ float wc[NC_][ND_][ND_];
  const int t = threadIdx.x;          // 0..127
  const int c = t >> 3, d = t & 7;
  const float* wd = weight + ((size_t)c * ND_ + d) * R_;
  for (int e = 0; e < ND_; ++e) {
    const float* we = weight + ((size_t)c * ND_ + e) * R_;
    float s = 0.f;
    for (int r = 0; r < R_; ++r) s = fmaf(wd[r], we[r], s);
    if (e == d) s += EPS_;
    wc[c][d][e] = s;
  }
  __syncthreads();
  if (t < NC_) {
    float a[8][16];
    for (int i = 0; i < 8; ++i)
      for (int j = 0; j < 8; ++j) {
        a[i][j] = wc[t][i][j];
        a[i][8 + j] = (i == j) ? 1.f : 0.f;
      }
    for (int k = 0; k < 8; ++k) {        // Gauss-Jordan (SPD + eps)
      const float piv = 1.0f / a[k][k];
      for (int j = 0; j < 16; ++j) a[k][j] *= piv;
      for (int i = 0; i < 8; ++i)
        if (i != k) {
          const float f = a[i][k];
          for (int j = 0; j < 16; ++j) a[i][j] -= f * a[k][j];
        }
    }
    for (int i = 0; i < 8; ++i)
      for (int j = 0; j < 8; ++j)
        sigma[t * 64 + i * 8 + j] = a[i][8 + j];
  }
}

// ---------------------------------------------------------------------------
// Kernel 2: per block b: proj = windows(x_b) @ weight^T via f32 WMMA with
// async double-buffered LDS staging; then q = p^T sigma p, out = mean sqrt(q).
// Each staged x slice {x[256w+k0+j]} serves K-range k0 (window slot w) AND
// K-range k0+256 (window slot (w+1) mod 128 == reference's mod-L wrap).
// ---------------------------------------------------------------------------
__global__ __launch_bounds__(256) void mccp_main(const float* __restrict__ x,
                                                 const float* __restrict__ weight,
                                                 const float* __restrict__ sigma,
                                                 float* __restrict__ out) {
  __shared__ __align__(16) unsigned char smem[SMEM_BYTES];
  float* sig     = (float*)(smem + SIG_OFF);
  float* proj    = (float*)(smem);            // epilogue overlay
  float* partbuf = (float*)(smem + PART_OFF); // epilogue overlay

  const int t    = threadIdx.x;
  const int lane = t & 31;
  const int wave = t >> 5;      // 8 waves
  const int b    = blockIdx.x;  // one block per batch row

  const int mrow = lane & 15;   // M (window) / N (cd) index within tile
  const int khi  = lane >> 4;   // K half: K = 4*s + 2*khi + v
  const int wb   = wave * 16;   // this wave's 16-window M-tile

  for (int i = t; i < NC_ * ND_ * ND_; i += 256) sig[i] = sigma[i];

  v8f acc[8] = {};  // 8 N-tiles of 16x16 f32

  // ---- async staging setup ------------------------------------------------
  const unsigned ldsBase =
      (unsigned)(unsigned long long)(__attribute__((address_space(3))) char*)smem;
  const int srow = t >> 1;           // 0..127: window row (xs) / cd row (wl)
  const int sseg = (t & 1) * 8;      // which 8 floats of the 16-float row
  const float* gx0 = x + (size_t)b * L_ + S_ * srow + sseg;  // never wraps: max 32767
  const float* gw0 = weight + (size_t)srow * R_ + sseg;
  const unsigned xsL = ldsBase + (unsigned)((srow * XSTR_ + sseg) << 2);
  const unsigned wlL = xsL + (unsigned)XS_BYTES;

  auto issue = [&](int k0, int bufsel) {
    const unsigned bo = (unsigned)(bufsel * BUF_BYTES);
    const float* gx = gx0 + k0;
    AISSUE(xsL + bo,       gx);
    AISSUE(xsL + bo + 16u, gx + 4);
    const float* gw = gw0 + k0;                 // weight half 0: k in [k0, k0+16)
    AISSUE(wlL + bo,       gw);
    AISSUE(wlL + bo + 16u, gw + 4);
    const float* gw2 = gw + S_;                 // weight half 1: k in [256+k0, ...)
    const unsigned wl2 = wlL + bo + (unsigned)(CD_ * XSTR_ * 4);
    AISSUE(wl2,        gw2);
    AISSUE(wl2 + 16u,  gw2 + 4);
  };

  const int arow0 = (wb + mrow) * XSTR_;
  const int arow1 = ((wb + mrow + 1) & (W_ - 1)) * XSTR_;  // mod-L wrap for free

  issue(0, 0);
  for (int c = 0; c < NCH_; ++c) {
    const int buf = c & 1;
    if (c + 1 < NCH_) {
      issue((c + 1) * KC_, buf ^ 1);
      asm volatile("s_wait_asynccnt 0x6" ::: "memory");   // chunk c's 6 ops landed
    } else {
      asm volatile("s_wait_asynccnt 0x0" ::: "memory");
    }
    __syncthreads();
    const float* xs  = (const float*)(smem + buf * BUF_BYTES);
    const float* wl0 = (const float*)(smem + buf * BUF_BYTES + XS_BYTES);
    const float* wl1 = wl0 + CD_ * XSTR_;
#pragma unroll
    for (int s = 0; s < KC_ / 4; ++s) {
      const int kl = 4 * s + 2 * khi;
      v2f a0 = *(const v2f*)(xs + arow0 + kl);
      v2f a1 = *(const v2f*)(xs + arow1 + kl);
#pragma unroll
      for (int n = 0; n < 8; ++n) {
        v2f b0 = *(const v2f*)(wl0 + (16 * n + mrow) * XSTR_ + kl);
        acc[n] = __builtin_amdgcn_wmma_f32_16x16x4_f32(
            false, a0, false, b0, (short)0, acc[n], false, false);
        v2f b1 = *(const v2f*)(wl1 + (16 * n + mrow) * XSTR_ + kl);
        acc[n] = __builtin_amdgcn_wmma_f32_16x16x4_f32(
            false, a1, false, b1, (short)0, acc[n], false, false);
      }
    }
    __syncthreads();
  }

  // ---- Epilogue: quad-form + sqrt + mean, two 64-window phases -----------
  float part = 0.f;
  const int qc = t & 15;   // c
  const int qw = t >> 4;   // 0..15 window sub-index
#pragma unroll 1
  for (int ph = 0; ph < 2; ++ph) {
    __syncthreads();
    if ((wave >> 2) == ph) {
      // D-tile layout: VGPR j holds (M = j + 8*khi, N = 16*n + mrow)
      const int prow0 = wb - ph * 64;
#pragma unroll
      for (int n = 0; n < 8; ++n) {
        const int cd = 16 * n + mrow;
#pragma unroll
        for (int j = 0; j < 8; ++j)
          proj[(prow0 + j + 8 * khi) * PSTR_ + cd] = acc[n][j];
      }
    }
    __syncthreads();
#pragma unroll
    for (int i = 0; i < 4; ++i) {
      const float* p = proj + (qw + 16 * i) * PSTR_ + qc * ND_;
      float q = 0.f;
#pragma unroll
      for (int d = 0; d < 8; ++d) {
        float sd = 0.f;
#pragma unroll
        for (int e = 0; e < 8; ++e) sd = fmaf(sig[qc * 64 + d * 8 + e], p[e], sd);
        q = fmaf(p[d], sd, q);
      }
      part += sqrtf(fmaxf(q, 0.f));
    }
  }
  partbuf[t] = part;
  __syncthreads();
  // Fixed-order reduction: deterministic across replays, no atomics.
  if (t < NC_) {
    float tot = 0.f;
#pragma unroll
    for (int i = 0; i < 16; ++i) tot += partbuf[t + 16 * i];
    out[b * NC_ + t] = tot * (1.0f / W_);
  }
}

// ---------------------------------------------------------------------------
extern "C" void kernel_launch(void* const* d_in, const int* in_sizes, int n_in,
                              void* d_out, int out_size, void* d_ws, size_t ws_size,
                              hipStream_t stream) {
  (void)in_sizes; (void)n_in; (void)out_size; (void)ws_size;
  const float* x      = (const float*)d_in[0];  // (512, 32768) f32
  const float* weight = (const float*)d_in[1];  // (16, 8, 512) f32
  float* out   = (float*)d_out;                 // (512, 16) f32
  float* sigma = (float*)d_ws;                  // 16*64 f32 scratch

  mccp_prep<<<1, 128, 0, stream>>>(weight, sigma);
  mccp_main<<<B_, 256, 0, stream>>>(x, weight, sigma, out);
}